// TempSoftPlus_20401094656133
// MI455X (gfx1250) — compile-verified
//
#include <hip/hip_runtime.h>
#include <hip/hip_bf16.h>
#include <math.h>

// ---------------------------------------------------------------------------
// TempSoftPlus for MI455X (gfx1250, wave32).
// out[n] = f( dinv[n]^2 * y[n] + sum_{e: col=n, row!=col} dinv[row]*dinv[col]*y[row] )
// with y[i] = x[i]·w  (Linear pushed through the linear aggregation),
// f(o) = 1/(softplus(o)+0.5) with inf guard.
// ---------------------------------------------------------------------------

typedef float v2f __attribute__((ext_vector_type(2)));
typedef float v8f __attribute__((ext_vector_type(8)));

#define D_DIM 128

// ---- 1. zero degree counters ----------------------------------------------
__global__ void k_zero_deg(unsigned* __restrict__ deg, int n) {
    int i = blockIdx.x * blockDim.x + threadIdx.x;
    if (i < n) deg[i] = 0u;
}

// ---- 2. degree accumulation (self-loops excluded) -------------------------
__global__ void k_degree(const int* __restrict__ ei, unsigned* __restrict__ deg,
                         int E) {
    int e = blockIdx.x * blockDim.x + threadIdx.x;
    if (e >= E) return;
    int r = ei[e];        // source
    int c = ei[E + e];    // target
    if (r != c) atomicAdd(&deg[c], 1u);
}

// ---- 3. y = x @ w  via V_WMMA_F32_16X16X4_F32 -----------------------------
// One wave of 32 lanes computes 16 rows. A-matrix (16x4 f32) layout per ISA:
//   lanes 0-15 : M = lane, VGPR0 = K0, VGPR1 = K1
//   lanes16-31 : M = lane-16, VGPR0 = K2, VGPR1 = K3
// B (4x16) replicates w[k..k+3] across all columns, so D[:,n] == y for all n.
__global__ void k_matvec_wmma(const float* __restrict__ x,
                              const float* __restrict__ w,
                              float* __restrict__ y,
                              int nTiles, int N) {
    __shared__ float wl[D_DIM];
    int tid = threadIdx.x;
    if (tid < D_DIM) wl[tid] = w[tid];
    __syncthreads();

    int lane = tid & 31;
    int wid  = blockIdx.x * (blockDim.x >> 5) + (tid >> 5);
    if (wid >= nTiles) return;                  // wave-uniform: EXEC stays all-1s

    int rowBase = wid << 4;
    int m  = lane & 15;
    int kh = lane >> 4;                         // 0 -> K%4 in {0,1}; 1 -> {2,3}
    int row = rowBase + m;
    if (row >= N) row = N - 1;                  // safe clamp (N%16==0 here anyway)

    const float* xr = x + (size_t)row * D_DIM + 2 * kh;
    const float* wr = wl + 2 * kh;

    v8f acc = {};
#pragma unroll
    for (int s = 0; s < D_DIM / 4; ++s) {
        v2f a, b;
        a.x = xr[4 * s];                        // 8B-aligned pair -> b64 load
        a.y = xr[4 * s + 1];
        b.x = wr[4 * s];
        b.y = wr[4 * s + 1];
        // (neg_a, A, neg_b, B, c_mod, C, reuse_a, reuse_b)
        acc = __builtin_amdgcn_wmma_f32_16x16x4_f32(
            false, a, false, b, (short)0, acc, false, false);
    }

    // D column 0: lane 0 holds M=0..7 in acc[0..7], lane 16 holds M=8..15.
    if (lane == 0) {
#pragma unroll
        for (int r = 0; r < 8; ++r)
            if (rowBase + r < N) y[rowBase + r] = acc[r];
    } else if (lane == 16) {
#pragma unroll
        for (int r = 0; r < 8; ++r)
            if (rowBase + 8 + r < N) y[rowBase + 8 + r] = acc[r];
    }
}

// ---- 4. dinv = rsqrt(deg+1); acc = dinv^2 * y (self-loop term) ------------
__global__ void k_node_init(const unsigned* __restrict__ deg,
                            const float* __restrict__ y,
                            float* __restrict__ dinv,
                            float* __restrict__ acc, int n) {
    int i = blockIdx.x * blockDim.x + threadIdx.x;
    if (i >= n) return;
    float d  = (float)deg[i] + 1.0f;            // +1 from added self loop
    float di = __frsqrt_rn(d);
    dinv[i] = di;
    acc[i]  = di * di * y[i];
}

// ---- 5. edge scatter of scalar messages -----------------------------------
__global__ void k_edge_scatter(const int* __restrict__ ei,
                               const float* __restrict__ dinv,
                               const float* __restrict__ y,
                               float* __restrict__ acc, int E) {
    int e = blockIdx.x * blockDim.x + threadIdx.x;
    if (e >= E) return;
    int r = ei[e];
    int c = ei[E + e];
    if (r == c) return;                         // removed self loops
    atomicAdd(&acc[c], dinv[r] * dinv[c] * y[r]);
}

// ---- 6. finalize: 1/(softplus(o)+tau0), inf -> 0 --------------------------
__global__ void k_finalize(const float* __restrict__ acc,
                           float* __restrict__ out, int n) {
    int i = blockIdx.x * blockDim.x + threadIdx.x;
    if (i >= n) return;
    float o  = acc[i];
    // stable softplus: max(o,0) + log1p(exp(-|o|))
    float sp = fmaxf(o, 0.0f) + log1pf(expf(-fabsf(o)));
    float t  = 1.0f / (sp + 0.5f);
    out[i] = isinf(t) ? 0.0f : t;
}

// ---------------------------------------------------------------------------
extern "C" void kernel_launch(void* const* d_in, const int* in_sizes, int n_in,
                              void* d_out, int out_size, void* d_ws, size_t ws_size,
                              hipStream_t stream) {
    const float* x   = (const float*)d_in[0];
    const int*   ei  = (const int*)d_in[1];
    const float* lw  = (const float*)d_in[2];
    float*       out = (float*)d_out;

    const int N = in_sizes[0] / D_DIM;
    const int E = in_sizes[1] / 2;

    // workspace layout: [deg u32 | dinv f32 | y f32 | acc f32], each N elems
    unsigned* deg  = (unsigned*)d_ws;
    float*    dinv = (float*)d_ws + N;
    float*    y    = (float*)d_ws + 2 * (size_t)N;
    float*    acc  = (float*)d_ws + 3 * (size_t)N;

    const int TB = 256;
    const int nBlkN = (N + TB - 1) / TB;
    const int nBlkE = (E + TB - 1) / TB;

    k_zero_deg<<<nBlkN, TB, 0, stream>>>(deg, N);
    k_degree<<<nBlkE, TB, 0, stream>>>(ei, deg, E);

    const int nTiles = (N + 15) / 16;                  // 3125 waves for N=50000
    const int wavesPerBlk = TB / 32;                   // 8
    k_matvec_wmma<<<(nTiles + wavesPerBlk - 1) / wavesPerBlk, TB, 0, stream>>>(
        x, lw, y, nTiles, N);

    k_node_init<<<nBlkN, TB, 0, stream>>>(deg, y, dinv, acc, N);
    k_edge_scatter<<<nBlkE, TB, 0, stream>>>(ei, dinv, y, acc, E);
    k_finalize<<<nBlkN, TB, 0, stream>>>(acc, out, N);
}